// GATNet_43130061586640
// MI455X (gfx1250) — compile-verified
//
#include <hip/hip_runtime.h>
#include <math.h>

typedef __attribute__((ext_vector_type(2))) float v2f;
typedef __attribute__((ext_vector_type(8))) float v8f;

#define TPB 256
#define NEG_SLOPE 0.2f
#define EPS 1e-16f

// ---------- helpers ----------
__device__ __forceinline__ float leaky(float v) {
    return v >= 0.0f ? v : NEG_SLOPE * v;
}
// monotonic float<->uint encoding so unsigned atomicMax == float max
__device__ __forceinline__ unsigned enc_f32(float f) {
    unsigned u = __float_as_uint(f);
    return (u & 0x80000000u) ? ~u : (u | 0x80000000u);
}
__device__ __forceinline__ float dec_f32(unsigned v) {
    unsigned u = (v & 0x80000000u) ? (v & 0x7FFFFFFFu) : ~v;
    return __uint_as_float(u);
}

// ---------- init fills ----------
__global__ void fill_u32(unsigned* __restrict__ p, unsigned v, int n) {
    int i = blockIdx.x * blockDim.x + threadIdx.x;
    if (i < n) p[i] = v;
}
__global__ void fill_f32(float* __restrict__ p, float v, int n) {
    int i = blockIdx.x * blockDim.x + threadIdx.x;
    if (i < n) p[i] = v;
}

// ---------- layer 1 feature transform: h1 = x @ W1  (N x 48 @ 48 x 8) ----------
// One wave per 16-row tile; 12 steps of V_WMMA_F32_16X16X4_F32; B padded to N=16.
// B padding done with a value-select (v_cndmask), NOT a predicated load, so the
// inner loop has no EXEC-mask branches.
__global__ void gemm_x_w1_wmma(const float* __restrict__ x,
                               const float* __restrict__ W,   // [48*8] row-major
                               float* __restrict__ h,         // [N*8]
                               int N, int ntiles) {
    int wave = (int)((blockIdx.x * blockDim.x + threadIdx.x) >> 5);
    int lane = threadIdx.x & 31;
    if (wave >= ntiles) return;               // wave-uniform guard; EXEC all-1 inside
    int row0  = wave * 16;
    int n     = lane & 15;                    // column index within tile
    int khalf = (lane >> 4) * 2;              // K offset: 0 for lo half, 2 for hi half
    int row   = row0 + n;                     // A-matrix row this lane feeds
    int rowc  = row < N ? row : N - 1;        // clamp for (generic) tail safety
    const float* xrow = x + (size_t)rowc * 48;

    float colmask = (n < 8) ? 1.0f : 0.0f;    // zero out padded B columns by value
    const float* wcol = W + (n & 7);          // always-valid address, no divergence

    v8f acc = {};
#pragma unroll
    for (int ks = 0; ks < 12; ++ks) {
        int kb = ks * 4 + khalf;
        v2f a; a.x = xrow[kb]; a.y = xrow[kb + 1];
        v2f b;
        b.x = wcol[(kb    ) * 8] * colmask;
        b.y = wcol[(kb + 1) * 8] * colmask;
        acc = __builtin_amdgcn_wmma_f32_16x16x4_f32(
            /*neg_a=*/false, a, /*neg_b=*/false, b,
            /*c_mod=*/(short)0, acc, /*reuse_a=*/false, /*reuse_b=*/false);
    }
    // D layout: VGPR v -> row (v + 8*(lane>=16)), col = lane&15
    int mh = (lane >> 4) * 8;
    if (n < 8) {
#pragma unroll
        for (int v = 0; v < 8; ++v) {
            int r = row0 + v + mh;
            if (r < N) h[(size_t)r * 8 + n] = acc[v];
        }
    }
}

// ---------- per-node attention logits: alpha_src/dst = h . a_src / a_dst ----------
template <int C>
__global__ void node_alpha(const float* __restrict__ h,
                           const float* __restrict__ a_src,
                           const float* __restrict__ a_dst,
                           float* __restrict__ asrc, float* __restrict__ adst, int N) {
    int i = blockIdx.x * blockDim.x + threadIdx.x;
    if (i >= N) return;
    const float* hr = h + (size_t)i * C;
    float s = 0.f, d = 0.f;
#pragma unroll
    for (int c = 0; c < C; ++c) { float v = hr[c]; s += v * a_src[c]; d += v * a_dst[c]; }
    asrc[i] = s; adst[i] = d;
}

// ---------- edge pass A: segment max of leaky_relu(as[src]+ad[dst]) over dst ----------
__global__ void edge_max(const long long* __restrict__ ei,   // [2*E] int64
                         const float* __restrict__ asrc,
                         const float* __restrict__ adst,
                         unsigned* __restrict__ menc, int E) {
    int e = blockIdx.x * blockDim.x + threadIdx.x;
    if (e >= E) return;
    __builtin_prefetch(ei + e + 8192, 0, 1);          // stream src ids (global_prefetch)
    __builtin_prefetch(ei + E + e + 8192, 0, 1);      // stream dst ids
    int s = (int)ei[e];
    int d = (int)ei[E + e];
    float v = leaky(asrc[s] + adst[d]);
    atomicMax(&menc[d], enc_f32(v));
}

// ---------- edge pass B: p = exp(e - m[dst]); z[dst]+=p; S[dst]+=p*h[src] ----------
// C == 8: h rows are 32B -> two float4 gathers.
__global__ void edge_acc8(const long long* __restrict__ ei,
                          const float* __restrict__ asrc,
                          const float* __restrict__ adst,
                          const unsigned* __restrict__ menc,
                          const float* __restrict__ h,
                          float* __restrict__ z, float* __restrict__ S, int E) {
    int e = blockIdx.x * blockDim.x + threadIdx.x;
    if (e >= E) return;
    __builtin_prefetch(ei + e + 8192, 0, 1);
    __builtin_prefetch(ei + E + e + 8192, 0, 1);
    int s = (int)ei[e];
    int d = (int)ei[E + e];
    float v = leaky(asrc[s] + adst[d]);
    float p = __expf(v - dec_f32(menc[d]));
    atomicAdd(&z[d], p);
    const float4* hs = (const float4*)(h + (size_t)s * 8);  // 16B-aligned
    float4 h0 = hs[0];
    float4 h1 = hs[1];
    float* Sd = S + (size_t)d * 8;
    atomicAdd(&Sd[0], p * h0.x); atomicAdd(&Sd[1], p * h0.y);
    atomicAdd(&Sd[2], p * h0.z); atomicAdd(&Sd[3], p * h0.w);
    atomicAdd(&Sd[4], p * h1.x); atomicAdd(&Sd[5], p * h1.y);
    atomicAdd(&Sd[6], p * h1.z); atomicAdd(&Sd[7], p * h1.w);
}

// C == 10: h rows are 40B (8B-aligned) -> five float2 gathers.
__global__ void edge_acc10(const long long* __restrict__ ei,
                           const float* __restrict__ asrc,
                           const float* __restrict__ adst,
                           const unsigned* __restrict__ menc,
                           const float* __restrict__ h,
                           float* __restrict__ z, float* __restrict__ S, int E) {
    int e = blockIdx.x * blockDim.x + threadIdx.x;
    if (e >= E) return;
    __builtin_prefetch(ei + e + 8192, 0, 1);
    __builtin_prefetch(ei + E + e + 8192, 0, 1);
    int s = (int)ei[e];
    int d = (int)ei[E + e];
    float v = leaky(asrc[s] + adst[d]);
    float p = __expf(v - dec_f32(menc[d]));
    atomicAdd(&z[d], p);
    const float2* hs = (const float2*)(h + (size_t)s * 10); // 8B-aligned
    float* Sd = S + (size_t)d * 10;
#pragma unroll
    for (int c = 0; c < 5; ++c) {
        float2 hv = hs[c];
        atomicAdd(&Sd[2 * c    ], p * hv.x);
        atomicAdd(&Sd[2 * c + 1], p * hv.y);
    }
}

// ---------- finalize layer 1: relu(S/(z+eps)+b1), then h2 = h @ W2 (8 -> 10) ----------
__global__ void finalize1(const float* __restrict__ S, const float* __restrict__ z,
                          const float* __restrict__ b1, const float* __restrict__ W2,
                          float* __restrict__ h2, int N) {
    int i = blockIdx.x * blockDim.x + threadIdx.x;
    if (i >= N) return;
    float inv = 1.0f / (z[i] + EPS);
    float hv[8];
#pragma unroll
    for (int c = 0; c < 8; ++c) {
        float v = S[(size_t)i * 8 + c] * inv + b1[c];
        hv[c] = v > 0.0f ? v : 0.0f;
    }
#pragma unroll
    for (int o = 0; o < 10; ++o) {
        float acc = 0.f;
#pragma unroll
        for (int c = 0; c < 8; ++c) acc += hv[c] * W2[c * 10 + o];
        h2[(size_t)i * 10 + o] = acc;
    }
}

// ---------- finalize layer 2: S2/(z2+eps)+b2 then log_softmax over 10 classes ----------
__global__ void finalize2(const float* __restrict__ S, const float* __restrict__ z,
                          const float* __restrict__ b, float* __restrict__ out, int N) {
    int i = blockIdx.x * blockDim.x + threadIdx.x;
    if (i >= N) return;
    float inv = 1.0f / (z[i] + EPS);
    float v[10];
    float mx = -INFINITY;
#pragma unroll
    for (int c = 0; c < 10; ++c) {
        v[c] = S[(size_t)i * 10 + c] * inv + b[c];
        mx = fmaxf(mx, v[c]);
    }
    float sum = 0.f;
#pragma unroll
    for (int c = 0; c < 10; ++c) sum += __expf(v[c] - mx);
    float lse = __logf(sum) + mx;
#pragma unroll
    for (int c = 0; c < 10; ++c) out[(size_t)i * 10 + c] = v[c] - lse;
}

extern "C" void kernel_launch(void* const* d_in, const int* in_sizes, int n_in,
                              void* d_out, int out_size, void* d_ws, size_t ws_size,
                              hipStream_t stream) {
    const float*     x      = (const float*)d_in[0];
    const long long* ei     = (const long long*)d_in[1];   // int64 edge_index [2,E]
    const float*     W1     = (const float*)d_in[3];
    const float*     a_src1 = (const float*)d_in[4];
    const float*     a_dst1 = (const float*)d_in[5];
    const float*     b1     = (const float*)d_in[6];
    const float*     W2     = (const float*)d_in[7];
    const float*     a_src2 = (const float*)d_in[8];
    const float*     a_dst2 = (const float*)d_in[9];
    const float*     b2     = (const float*)d_in[10];
    const int N = in_sizes[0] / 48;
    const int E = in_sizes[1] / 2;
    float* out = (float*)d_out;

    // workspace layout (44 floats per node ~= 17.6 MB)
    float*    h1  = (float*)d_ws;                  // 8N
    float*    as1 = h1 + (size_t)8 * N;            // N
    float*    ad1 = as1 + N;                       // N
    unsigned* m1  = (unsigned*)(ad1 + N);          // N
    float*    z1  = (float*)(m1 + N);              // N   (z1,S1 contiguous for one fill)
    float*    S1  = z1 + N;                        // 8N
    float*    h2  = S1 + (size_t)8 * N;            // 10N
    float*    as2 = h2 + (size_t)10 * N;           // N
    float*    ad2 = as2 + N;                       // N
    unsigned* m2  = (unsigned*)(ad2 + N);          // N
    float*    z2  = (float*)(m2 + N);              // N   (z2,S2 contiguous)
    float*    S2  = z2 + N;                        // 10N

    const unsigned ENC_NEG_INF = 0x007FFFFFu;      // enc_f32(-inf)
    int nodeB = (N + TPB - 1) / TPB;
    int edgeB = (E + TPB - 1) / TPB;
    int tiles = (N + 15) / 16;
    int gemmB = (tiles + 7) / 8;                   // 8 waves (256 thr) per block

    // ---- layer 1 ----
    gemm_x_w1_wmma<<<gemmB, TPB, 0, stream>>>(x, W1, h1, N, tiles);
    node_alpha<8><<<nodeB, TPB, 0, stream>>>(h1, a_src1, a_dst1, as1, ad1, N);
    fill_u32<<<nodeB, TPB, 0, stream>>>(m1, ENC_NEG_INF, N);
    fill_f32<<<(9 * N + TPB - 1) / TPB, TPB, 0, stream>>>(z1, 0.0f, 9 * N);
    edge_max<<<edgeB, TPB, 0, stream>>>(ei, as1, ad1, m1, E);
    edge_acc8<<<edgeB, TPB, 0, stream>>>(ei, as1, ad1, m1, h1, z1, S1, E);
    finalize1<<<nodeB, TPB, 0, stream>>>(S1, z1, b1, W2, h2, N);

    // ---- layer 2 ----
    node_alpha<10><<<nodeB, TPB, 0, stream>>>(h2, a_src2, a_dst2, as2, ad2, N);
    fill_u32<<<nodeB, TPB, 0, stream>>>(m2, ENC_NEG_INF, N);
    fill_f32<<<(11 * N + TPB - 1) / TPB, TPB, 0, stream>>>(z2, 0.0f, 11 * N);
    edge_max<<<edgeB, TPB, 0, stream>>>(ei, as2, ad2, m2, E);
    edge_acc10<<<edgeB, TPB, 0, stream>>>(ei, as2, ad2, m2, h2, z2, S2, E);
    finalize2<<<nodeB, TPB, 0, stream>>>(S2, z2, b2, out, N);
}